// AttentivePool_60361470378678
// MI455X (gfx1250) — compile-verified
//
#include <hip/hip_runtime.h>
#include <math.h>
#include <stdint.h>

typedef __attribute__((ext_vector_type(2))) float v2f;
typedef __attribute__((ext_vector_type(8))) float v8f;

#define BB 16
#define CC 1536
#define TT 2000
#define AT 128   // ATTN

// ---------------------------------------------------------------------------
// Async global->LDS copy (CDNA5 GLOBAL_LOAD_ASYNC_TO_LDS_B128, ASYNCcnt
// tracked). Builtin signature (from hipcc diagnostic): param0 is a pointer to
// int __attribute__((vector_size(16))) in addrspace(1); param1 addrspace(3).
// Falls back to a plain 16-byte copy when the builtin is absent.
// ---------------------------------------------------------------------------
#if __has_builtin(__builtin_amdgcn_global_load_async_to_lds_b128)
#define HAVE_ASYNC_LDS 1
#endif

typedef int v4i_gcc __attribute__((vector_size(16)));
typedef __attribute__((address_space(1))) v4i_gcc* g4p_t;  // global int4*
typedef __attribute__((address_space(3))) v4i_gcc* l4p_t;  // LDS    int4*

__device__ __forceinline__ void copy16_g2l(const float* g, float* l) {
#ifdef HAVE_ASYNC_LDS
    __builtin_amdgcn_global_load_async_to_lds_b128(
        (g4p_t)(uintptr_t)g, (l4p_t)(uintptr_t)l, 0, 0);
#else
    *(float4*)l = *(const float4*)g;
#endif
}

__device__ __forceinline__ void copy_join() {
#ifdef HAVE_ASYNC_LDS
  #if __has_builtin(__builtin_amdgcn_s_wait_asynccnt)
    __builtin_amdgcn_s_wait_asynccnt(0);
  #else
    asm volatile("s_wait_asynccnt 0" ::: "memory");
  #endif
#endif
}

// ---------------------------------------------------------------------------
// Kernel 0: per-batch mask sums  msum[b] = sum_t mask[b,t]
// ---------------------------------------------------------------------------
__global__ void k_masksum(const int* __restrict__ mask, float* __restrict__ msum) {
    __shared__ float s[256];
    int b = blockIdx.x, tid = threadIdx.x;
    float acc = 0.f;
    for (int t = tid; t < TT; t += 256) acc += (float)mask[b * TT + t];
    s[tid] = acc; __syncthreads();
    for (int off = 128; off; off >>= 1) {
        if (tid < off) s[tid] += s[tid + off];
        __syncthreads();
    }
    if (tid == 0) msum[b] = s[0];
}

// ---------------------------------------------------------------------------
// Kernel 1: masked weighted mean/std per (b,c) over T  (reads x once)
// ---------------------------------------------------------------------------
__global__ void k_stats(const float* __restrict__ x, const int* __restrict__ mask,
                        const float* __restrict__ msum,
                        float* __restrict__ mu, float* __restrict__ sd) {
    __shared__ float s1[256], s2[256];
    int c = blockIdx.x, b = blockIdx.y, tid = threadIdx.x;
    const float* xr = x + ((size_t)b * CC + c) * TT;
    const int*   mr = mask + b * TT;
    float a1 = 0.f, a2 = 0.f;
    for (int t = tid; t < TT; t += 256) {
        float m = (float)mr[t];
        float v = xr[t];
        a1 += v * m;
        a2 += v * v * m;
    }
    s1[tid] = a1; s2[tid] = a2; __syncthreads();
    for (int off = 128; off; off >>= 1) {
        if (tid < off) { s1[tid] += s1[tid + off]; s2[tid] += s2[tid + off]; }
        __syncthreads();
    }
    if (tid == 0) {
        float inv = 1.f / msum[b];
        float m1 = s1[0] * inv, m2 = s2[0] * inv;
        float var = m2 - m1 * m1;
        mu[b * CC + c] = m1;
        sd[b * CC + c] = sqrtf(fmaxf(var, 1e-9f));
    }
}

// ---------------------------------------------------------------------------
// Kernel 2: fold broadcast mu/sd halves of feat through w1 into a per-(b,o)
// bias:  beff[b,o] = b1[o] + w1[o,C:2C].mu[b] + w1[o,2C:3C].sd[b]
// ---------------------------------------------------------------------------
__global__ void k_bias(const float* __restrict__ w1, const float* __restrict__ b1,
                       const float* __restrict__ mu, const float* __restrict__ sd,
                       float* __restrict__ beff) {
    __shared__ float s[256];
    int o = blockIdx.x, b = blockIdx.y, tid = threadIdx.x;
    const float* wm = w1 + (size_t)o * (3 * CC) + CC;
    const float* ws = wm + CC;
    float acc = 0.f;
    for (int c = tid; c < CC; c += 256)
        acc += wm[c] * mu[b * CC + c] + ws[c] * sd[b * CC + c];
    s[tid] = acc; __syncthreads();
    for (int off = 128; off; off >>= 1) {
        if (tid < off) s[tid] += s[tid + off];
        __syncthreads();
    }
    if (tid == 0) beff[b * AT + o] = b1[o] + s[0];
}

// ---------------------------------------------------------------------------
// Kernel 3: GEMM1 (w1[:, :C] @ x[b]) via V_WMMA_F32_16X16X4_F32, fused with
// channel LayerNorm + ReLU + tanh. One block = (b, 16-wide t tile); 8 waves,
// wave w owns output rows 16w..16w+15. K streamed through LDS in 64-chunks,
// staged with async global->LDS B128 transfers.
//   A tile w1s: [m][kk], stride 68  -> float2 fragment gather, banks 4*lane
//   B tile xs:  [kk][tt], stride 20 -> 16B-contiguous rows for B128 copies;
//                gather banks lanes0-15 [c..c+15], lanes16-31 [c+40..c+55]
// ---------------------------------------------------------------------------
__global__ void __launch_bounds__(256)
k_gemm1_ln(const float* __restrict__ x, const float* __restrict__ w1,
           const float* __restrict__ beff, const float* __restrict__ ln_g,
           const float* __restrict__ ln_b, float* __restrict__ th) {
    __shared__ float w1s[AT * 68];   // [m][kk]
    __shared__ float xs[64 * 20];    // [kk][tt]
    __shared__ float hs[AT * 17];    // h tile for cross-wave LayerNorm
    __shared__ float cmu[16], crs[16];

    int tid = threadIdx.x;
    int wave = tid >> 5, lane = tid & 31;
    int l15 = lane & 15, hi = lane >> 4;      // hi: K+2/K+3 half (A/B), M+8 half (C/D)
    int b  = blockIdx.y;
    int t0 = blockIdx.x * 16;
    int m0 = wave * 16;

    const float* xb = x + ((size_t)b * CC) * TT + t0;
    v8f acc = {};

    for (int k0 = 0; k0 < CC; k0 += 64) {
        __syncthreads();
        // stage w1 chunk: 128 rows x 64 k = 2048 x 16B transfers
        for (int idx = tid; idx < AT * 16; idx += 256) {
            int m = idx >> 4, q = (idx & 15) << 2;
            copy16_g2l(&w1[(size_t)m * (3 * CC) + k0 + q], &w1s[m * 68 + q]);
        }
        // stage x chunk: 64 channels x 16 t = 256 x 16B transfers (1/thread)
        {
            int kk = tid >> 2, q = (tid & 3) << 2;
            copy16_g2l(&xb[(size_t)(k0 + kk) * TT + q], &xs[kk * 20 + q]);
        }
        copy_join();
        __syncthreads();

        #pragma unroll
        for (int kk = 0; kk < 64; kk += 4) {
            v2f af, bf;
            const float* ap = &w1s[(m0 + l15) * 68 + kk + 2 * hi];
            af.x = ap[0]; af.y = ap[1];           // A 16x4: lanes<16 K0/K1, lanes>=16 K2/K3
            const float* bp = &xs[(kk + 2 * hi) * 20 + l15];
            bf.x = bp[0]; bf.y = bp[20];          // B 4x16: rows kk(+2) / kk+1(+3), N=lane&15
            acc = __builtin_amdgcn_wmma_f32_16x16x4_f32(false, af, false, bf,
                                                        (short)0, acc, false, false);
        }
    }

    // scatter C/D fragment (+bias) into LDS for the cross-wave LayerNorm
    #pragma unroll
    for (int r = 0; r < 8; ++r) {
        int m = m0 + r + 8 * hi;                  // C/D: VGPR r -> rows r and r+8
        hs[m * 17 + l15] = acc[r] + beff[b * AT + m];
    }
    __syncthreads();

    // LayerNorm over 128 channels per column t
    if (tid < 16) {
        float s = 0.f, s2 = 0.f;
        for (int m = 0; m < AT; ++m) { float v = hs[m * 17 + tid]; s += v; s2 += v * v; }
        float mean = s * (1.f / AT);
        float var  = s2 * (1.f / AT) - mean * mean;
        cmu[tid] = mean;
        crs[tid] = rsqrtf(var + 1e-5f);
    }
    __syncthreads();

    for (int idx = tid; idx < AT * 16; idx += 256) {
        int m = idx >> 4, tt = idx & 15;
        float v = (hs[m * 17 + tt] - cmu[tt]) * crs[tt] * ln_g[m] + ln_b[m];
        v = fmaxf(v, 0.f);
        th[((size_t)b * AT + m) * TT + t0 + tt] = tanhf(v);
    }
}

// ---------------------------------------------------------------------------
// Kernel 4: GEMM2  a[b] = w2 @ tanh(h)[b] + b2, same WMMA tiling.
// Block = (t tile, 128-row c tile, b); 8 waves x 16x16 tiles, K=128 in 2 chunks.
// ---------------------------------------------------------------------------
__global__ void __launch_bounds__(256)
k_gemm2(const float* __restrict__ th, const float* __restrict__ w2,
        const float* __restrict__ b2, float* __restrict__ amat) {
    __shared__ float w2s[128 * 68];  // [c][kk]
    __shared__ float ths[64 * 20];   // [kk][tt]

    int tid = threadIdx.x;
    int wave = tid >> 5, lane = tid & 31;
    int l15 = lane & 15, hi = lane >> 4;
    int t0 = blockIdx.x * 16;
    int c0 = blockIdx.y * 128;
    int b  = blockIdx.z;
    int m0 = wave * 16;

    v8f acc = {};
    for (int k0 = 0; k0 < AT; k0 += 64) {
        __syncthreads();
        // stage w2 chunk: 128 rows x 64 k = 2048 x 16B transfers
        for (int idx = tid; idx < 128 * 16; idx += 256) {
            int c = idx >> 4, q = (idx & 15) << 2;
            copy16_g2l(&w2[(size_t)(c0 + c) * AT + k0 + q], &w2s[c * 68 + q]);
        }
        // stage tanh(h) chunk: 64 k x 16 t = 256 x 16B transfers (1/thread)
        {
            int kk = tid >> 2, q = (tid & 3) << 2;
            copy16_g2l(&th[((size_t)b * AT + k0 + kk) * TT + t0 + q],
                       &ths[kk * 20 + q]);
        }
        copy_join();
        __syncthreads();

        #pragma unroll
        for (int kk = 0; kk < 64; kk += 4) {
            v2f af, bf;
            const float* ap = &w2s[(m0 + l15) * 68 + kk + 2 * hi];
            af.x = ap[0]; af.y = ap[1];
            const float* bp = &ths[(kk + 2 * hi) * 20 + l15];
            bf.x = bp[0]; bf.y = bp[20];
            acc = __builtin_amdgcn_wmma_f32_16x16x4_f32(false, af, false, bf,
                                                        (short)0, acc, false, false);
        }
    }
    #pragma unroll
    for (int r = 0; r < 8; ++r) {
        int c = c0 + m0 + r + 8 * hi;
        amat[((size_t)b * CC + c) * TT + t0 + l15] = acc[r] + b2[c];
    }
}

// ---------------------------------------------------------------------------
// Kernel 5: single-pass online masked softmax over T fused with the weighted
// mean/std pooling. Running (max M, S=sum e, P=sum e*x, Q=sum e*x^2) with
// rescaling; mask==0 frames contribute nothing (equivalent to -inf logits).
// ---------------------------------------------------------------------------
__global__ void k_pool(const float* __restrict__ x, const float* __restrict__ amat,
                       const int* __restrict__ mask, float* __restrict__ out) {
    __shared__ float sM[256], sS[256], sP[256], sQ[256];
    int c = blockIdx.x, b = blockIdx.y, tid = threadIdx.x;
    const float* ar = amat + ((size_t)b * CC + c) * TT;
    const float* xr = x    + ((size_t)b * CC + c) * TT;
    const int*   mr = mask + b * TT;

    float M = -3.4e38f, S = 0.f, P = 0.f, Q = 0.f;
    for (int t = tid; t < TT; t += 256) {
        if (mr[t]) {
            float av = ar[t], xv = xr[t];
            if (av > M) {
                float r = __expf(M - av);
                S *= r; P *= r; Q *= r; M = av;
            }
            float e = __expf(av - M);
            S += e; P += e * xv; Q += e * xv * xv;
        }
    }
    sM[tid] = M; sS[tid] = S; sP[tid] = P; sQ[tid] = Q;
    __syncthreads();
    for (int off = 128; off; off >>= 1) {
        if (tid < off) {
            float S2 = sS[tid + off];
            if (S2 > 0.f) {
                float M2 = sM[tid + off], P2 = sP[tid + off], Q2 = sQ[tid + off];
                float M1 = sM[tid], S1 = sS[tid], P1 = sP[tid], Q1 = sQ[tid];
                if (S1 <= 0.f) {
                    sM[tid] = M2; sS[tid] = S2; sP[tid] = P2; sQ[tid] = Q2;
                } else {
                    float Mn = fmaxf(M1, M2);
                    float r1 = __expf(M1 - Mn), r2 = __expf(M2 - Mn);
                    sM[tid] = Mn;
                    sS[tid] = S1 * r1 + S2 * r2;
                    sP[tid] = P1 * r1 + P2 * r2;
                    sQ[tid] = Q1 * r1 + Q2 * r2;
                }
            }
        }
        __syncthreads();
    }
    if (tid == 0) {
        float inv = 1.f / sS[0];
        float m1  = sP[0] * inv;
        float var = sQ[0] * inv - m1 * m1;
        out[(size_t)b * (2 * CC) + c]      = m1;                       // mu2
        out[(size_t)b * (2 * CC) + CC + c] = sqrtf(fmaxf(var, 1e-9f)); // sd2
    }
}

// ---------------------------------------------------------------------------
extern "C" void kernel_launch(void* const* d_in, const int* in_sizes, int n_in,
                              void* d_out, int out_size, void* d_ws, size_t ws_size,
                              hipStream_t stream) {
    const float* x    = (const float*)d_in[0];   // [B,C,T]
    const int*   mask = (const int*)  d_in[1];   // [B,1,T]
    const float* w1   = (const float*)d_in[2];   // [128, 3C]
    const float* b1   = (const float*)d_in[3];   // [128]
    const float* ln_g = (const float*)d_in[4];   // [128]
    const float* ln_b = (const float*)d_in[5];   // [128]
    const float* w2   = (const float*)d_in[6];   // [C, 128]
    const float* b2   = (const float*)d_in[7];   // [C]
    float* out = (float*)d_out;                  // [B, 2C]

    float* ws   = (float*)d_ws;
    float* msum = ws;                                 // 16
    float* mu   = msum + BB;                          // B*C
    float* sd   = mu + (size_t)BB * CC;               // B*C
    float* beff = sd + (size_t)BB * CC;               // B*128
    float* th   = beff + (size_t)BB * AT;             // B*128*T   (tanh(h))
    float* amat = th + (size_t)BB * AT * TT;          // B*C*T     (logits)

    k_masksum<<<BB, 256, 0, stream>>>(mask, msum);
    k_stats<<<dim3(CC, BB), 256, 0, stream>>>(x, mask, msum, mu, sd);
    k_bias<<<dim3(AT, BB), 256, 0, stream>>>(w1, b1, mu, sd, beff);
    k_gemm1_ln<<<dim3(TT / 16, BB), 256, 0, stream>>>(x, w1, beff, ln_g, ln_b, th);
    k_gemm2<<<dim3(TT / 16, CC / 128, BB), 256, 0, stream>>>(th, w2, b2, amat);
    k_pool<<<dim3(CC, BB), 256, 0, stream>>>(x, amat, mask, out);
}